// SetAbstractionLayer_66297115181378
// MI455X (gfx1250) — compile-verified
//
#include <hip/hip_runtime.h>

typedef float v2f __attribute__((ext_vector_type(2)));
typedef float v8f __attribute__((ext_vector_type(8)));

#define NP        65536
#define NCENT     2048
#define KNN       32
#define RAD2      0.0025f
#define IN_DIM    40
#define H1D       64
#define H2D       128
#define OUTD      128
#define NROWS     (NCENT * KNN)   // 65536
#define BN_EPS    1e-5f
#define PI_F      3.14159265358979323846f

// ---------------------------------------------------------------------------
// Phase 1: Furthest Point Sampling. One persistent 1024-thread block.
// dist[] lives in global ws; each entry is read/written ONLY by its owning
// thread (p = tid + i*1024), so no cross-thread visibility issues.
// ---------------------------------------------------------------------------
__global__ __launch_bounds__(1024) void fps_kernel(
    const float* __restrict__ points, float* __restrict__ dist,
    float* __restrict__ cents /* [2048,2] */)
{
    const int tid = threadIdx.x;
    __shared__ float swv[32];
    __shared__ int   swi[32];
    __shared__ int   scur;

    for (int i = tid; i < NP; i += 1024) dist[i] = 3.0e38f;

    int cur = 0;
    if (tid == 0) { cents[0] = points[0]; cents[1] = points[1]; }

    for (int step = 1; step < NCENT; ++step) {
        const float cx = points[2 * cur];
        const float cy = points[2 * cur + 1];
        float best = -1.0f; int bidx = 0;
        #pragma unroll 4
        for (int i = 0; i < NP / 1024; ++i) {
            const int p = tid + (i << 10);
            const float dx = points[2 * p]     - cx;
            const float dy = points[2 * p + 1] - cy;
            const float d  = sqrtf(dx * dx + dy * dy + 1e-12f);
            float nd = fminf(dist[p], d);
            // lazy application of reference's dist[nxt] = 0 (cur chosen by argmax)
            if (p == cur && step != 1) nd = 0.0f;
            dist[p] = nd;
            if (nd > best) { best = nd; bidx = p; }
        }
        // wave-level argmax (tie -> smaller index, matching jnp.argmax)
        #pragma unroll
        for (int off = 16; off > 0; off >>= 1) {
            float ov = __shfl_xor(best, off, 32);
            int   oi = __shfl_xor(bidx, off, 32);
            if (ov > best || (ov == best && oi < bidx)) { best = ov; bidx = oi; }
        }
        const int wid = tid >> 5;
        if ((tid & 31) == 0) { swv[wid] = best; swi[wid] = bidx; }
        __syncthreads();
        if (wid == 0) {
            best = swv[tid & 31]; bidx = swi[tid & 31];
            #pragma unroll
            for (int off = 16; off > 0; off >>= 1) {
                float ov = __shfl_xor(best, off, 32);
                int   oi = __shfl_xor(bidx, off, 32);
                if (ov > best || (ov == best && oi < bidx)) { best = ov; bidx = oi; }
            }
            if (tid == 0) {
                scur = bidx;
                cents[2 * step]     = points[2 * bidx];
                cents[2 * step + 1] = points[2 * bidx + 1];
            }
        }
        __syncthreads();
        cur = scur;
        __syncthreads();
    }
}

// ---------------------------------------------------------------------------
// Phase 2: ball query (first-K in index order, pad with first hit) + Fourier
// encode. One wave per centroid.
// ---------------------------------------------------------------------------
__global__ __launch_bounds__(128) void ballq_encode_kernel(
    const float* __restrict__ points, const float* __restrict__ cents,
    float* __restrict__ feats /* [NROWS,40] */)
{
    const int lane = threadIdx.x & 31;
    const int wib  = threadIdx.x >> 5;
    const int m    = blockIdx.x * 4 + wib;   // centroid id, 2048 waves total

    __shared__ int sidx[4][KNN];
    int* myidx = sidx[wib];

    const float cx = cents[2 * m], cy = cents[2 * m + 1];
    const float c2 = cx * cx + cy * cy;

    int cnt = 0;
    for (int base = 0; base < NP && cnt < KNN; base += 32) {
        const int p = base + lane;
        const float px = points[2 * p], py = points[2 * p + 1];
        // match reference formula: c2 + p2 - 2*(c.p)
        const float d2 = c2 + px * px + py * py - 2.0f * (cx * px + cy * py);
        const bool hit = d2 <= RAD2;
        const unsigned mask = (unsigned)__ballot(hit);
        const int prefix = __popc(mask & ((1u << lane) - 1u));
        if (hit && (cnt + prefix) < KNN) myidx[cnt + prefix] = p;
        cnt += __popc(mask);
    }
    if (cnt > KNN) cnt = KNN;
    const int first = myidx[0];              // centroid itself guarantees >=1 hit
    const int pid   = (lane < cnt) ? myidx[lane] : first;

    const float rx = points[2 * pid]     - cx;
    const float ry = points[2 * pid + 1] - cy;

    float* f = feats + ((size_t)m * KNN + lane) * IN_DIM;
    float freq = PI_F;
    #pragma unroll
    for (int e = 0; e < 10; ++e) {
        const float ax = rx * freq, ay = ry * freq;
        f[4 * e + 0] = sinf(ax);
        f[4 * e + 1] = cosf(ax);
        f[4 * e + 2] = sinf(ay);
        f[4 * e + 3] = cosf(ay);
        freq *= 2.0f;
    }
}

// ---------------------------------------------------------------------------
// fp32 WMMA GEMM: Y[NROWS,NDIM] = X[NROWS,KDIM] @ W[KDIM,NDIM] + bias.
// One 16x16 output tile per wave via V_WMMA_F32_16X16X4_F32 (K-steps of 4).
// ---------------------------------------------------------------------------
template <int KDIM, int NDIM>
__global__ __launch_bounds__(256) void gemm_wmma_kernel(
    const float* __restrict__ X, const float* __restrict__ W,
    const float* __restrict__ bias, float* __restrict__ Y)
{
    const int lane  = threadIdx.x & 31;
    const int wave  = threadIdx.x >> 5;
    const int tilesN = NDIM / 16;
    const int tile  = blockIdx.x * 8 + wave;
    const int tm    = tile / tilesN;
    const int tn    = tile - tm * tilesN;

    const int row   = tm * 16 + (lane & 15);   // A fragment row
    const int col   = tn * 16 + (lane & 15);   // B/C fragment col
    const int khalf = (lane >> 4) * 2;         // K split across lane halves

    v8f c = {};
    #pragma unroll
    for (int k0 = 0; k0 < KDIM; k0 += 4) {
        const int kb = k0 + khalf;
        v2f a, b;
        a.x = X[(size_t)row * KDIM + kb];
        a.y = X[(size_t)row * KDIM + kb + 1];
        b.x = W[(size_t)kb * NDIM + col];
        b.y = W[(size_t)(kb + 1) * NDIM + col];
        c = __builtin_amdgcn_wmma_f32_16x16x4_f32(false, a, false, b,
                                                  (short)0, c, false, false);
    }

    const float bv = bias[col];
    const int rbase = tm * 16 + 8 * (lane >> 4);
    #pragma unroll
    for (int v = 0; v < 8; ++v)
        Y[(size_t)(rbase + v) * NDIM + col] = c[v] + bv;
}

// ---------------------------------------------------------------------------
// Deterministic two-stage BatchNorm statistics (no atomics).
// Stage 1: 256 blocks, each reduces 256 rows -> per-block column partials.
// ---------------------------------------------------------------------------
template <int NDIM>
__global__ __launch_bounds__(256) void colsum_partial_kernel(
    const float* __restrict__ Y, float* __restrict__ psum, float* __restrict__ psq)
{
    const int RP  = 256 / NDIM;
    const int t   = threadIdx.x;
    const int col = t & (NDIM - 1);
    const int rl  = t / NDIM;
    const int r0  = blockIdx.x * 256;

    float s = 0.0f, q = 0.0f;
    for (int r = r0 + rl; r < r0 + 256; r += RP) {
        const float v = Y[(size_t)r * NDIM + col];
        s += v; q += v * v;
    }
    __shared__ float ls[256], lq[256];
    ls[t] = s; lq[t] = q;
    __syncthreads();
    if (t < NDIM) {
        float ts = 0.0f, tq = 0.0f;
        #pragma unroll
        for (int i = 0; i < RP; ++i) { ts += ls[t + i * NDIM]; tq += lq[t + i * NDIM]; }
        psum[blockIdx.x * NDIM + t] = ts;
        psq [blockIdx.x * NDIM + t] = tq;
    }
}

// Stage 2: single block folds 256 partials per column -> mean, rstd.
template <int NDIM>
__global__ __launch_bounds__(128) void bn_finalize_kernel(
    const float* __restrict__ psum, const float* __restrict__ psq,
    float* __restrict__ meanv, float* __restrict__ rstdv)
{
    const int t = threadIdx.x;
    if (t < NDIM) {
        float s = 0.0f, q = 0.0f;
        for (int b = 0; b < 256; ++b) { s += psum[b * NDIM + t]; q += psq[b * NDIM + t]; }
        const float mean = s / (float)NROWS;
        const float var  = q / (float)NROWS - mean * mean;   // biased, matches jnp.var
        meanv[t] = mean;
        rstdv[t] = rsqrtf(var + BN_EPS);
    }
}

// BN affine + ReLU, in place.
template <int NDIM>
__global__ __launch_bounds__(256) void bn_relu_kernel(
    float* __restrict__ Y, const float* __restrict__ g, const float* __restrict__ be,
    const float* __restrict__ meanv, const float* __restrict__ rstdv)
{
    const size_t i  = (size_t)blockIdx.x * 256 + threadIdx.x;
    const int    c  = (int)(i & (NDIM - 1));
    const float  y  = Y[i];
    const float  h  = g[c] * (y - meanv[c]) * rstdv[c] + be[c];
    Y[i] = fmaxf(h, 0.0f);
}

// ---------------------------------------------------------------------------
// Phase 3 tail: GEMM3 (128x128) fused with maxpool over K=32 rows.
// One wave handles one (group, 16-col tile): two 16-row WMMA tiles, then
// max over 8 accum VGPRs and across the lane-16 halves.
// ---------------------------------------------------------------------------
__global__ __launch_bounds__(256) void gemm3_maxpool_kernel(
    const float* __restrict__ X, const float* __restrict__ W,
    const float* __restrict__ bias, float* __restrict__ out /* [2048,128] */)
{
    const int lane = threadIdx.x & 31;
    const int wave = threadIdx.x >> 5;
    const int tile = blockIdx.x * 8 + wave;     // 2048 groups * 8 col-tiles
    const int g    = tile >> 3;
    const int tn   = tile & 7;

    const int col   = tn * 16 + (lane & 15);
    const int khalf = (lane >> 4) * 2;
    const int r0    = g * 32 + (lane & 15);
    const int r1    = r0 + 16;

    v8f c0 = {}, c1 = {};
    #pragma unroll
    for (int k0 = 0; k0 < H2D; k0 += 4) {
        const int kb = k0 + khalf;
        v2f b, a0, a1;
        b.x  = W[(size_t)kb * OUTD + col];
        b.y  = W[(size_t)(kb + 1) * OUTD + col];
        a0.x = X[(size_t)r0 * H2D + kb];
        a0.y = X[(size_t)r0 * H2D + kb + 1];
        a1.x = X[(size_t)r1 * H2D + kb];
        a1.y = X[(size_t)r1 * H2D + kb + 1];
        c0 = __builtin_amdgcn_wmma_f32_16x16x4_f32(false, a0, false, b,
                                                   (short)0, c0, false, false);
        c1 = __builtin_amdgcn_wmma_f32_16x16x4_f32(false, a1, false, b,
                                                   (short)0, c1, false, false);
    }
    const float bv = bias[col];
    float mx = -3.0e38f;
    #pragma unroll
    for (int v = 0; v < 8; ++v) {
        mx = fmaxf(mx, c0[v] + bv);
        mx = fmaxf(mx, c1[v] + bv);
    }
    mx = fmaxf(mx, __shfl_xor(mx, 16, 32));     // combine the two row-halves
    if (lane < 16) out[(size_t)g * OUTD + tn * 16 + lane] = mx;
}

// ---------------------------------------------------------------------------
extern "C" void kernel_launch(void* const* d_in, const int* in_sizes, int n_in,
                              void* d_out, int out_size, void* d_ws, size_t ws_size,
                              hipStream_t stream) {
    const float* points = (const float*)d_in[0];
    const float* W1  = (const float*)d_in[1];
    const float* b1  = (const float*)d_in[2];
    const float* g1  = (const float*)d_in[3];
    const float* be1 = (const float*)d_in[4];
    const float* W2  = (const float*)d_in[5];
    const float* b2  = (const float*)d_in[6];
    const float* g2  = (const float*)d_in[7];
    const float* be2 = (const float*)d_in[8];
    const float* W3  = (const float*)d_in[9];
    const float* b3  = (const float*)d_in[10];

    float* out   = (float*)d_out;                     // [2048,128]
    float* cents = out + (size_t)NCENT * OUTD;        // [2048,2]

    float* ws = (float*)d_ws;
    size_t o = 0;
    float* dist  = ws + o; o += NP;                   // 65536
    float* feats = ws + o; o += (size_t)NROWS * IN_DIM;
    float* y1    = ws + o; o += (size_t)NROWS * H1D;
    float* y2    = ws + o; o += (size_t)NROWS * H2D;
    float* psum1 = ws + o; o += 256 * H1D;
    float* psq1  = ws + o; o += 256 * H1D;
    float* psum2 = ws + o; o += 256 * H2D;
    float* psq2  = ws + o; o += 256 * H2D;
    float* mean1 = ws + o; o += H1D;
    float* rstd1 = ws + o; o += H1D;
    float* mean2 = ws + o; o += H2D;
    float* rstd2 = ws + o; o += H2D;

    // 1) FPS -> centroids
    fps_kernel<<<1, 1024, 0, stream>>>(points, dist, cents);
    // 2) ball query + positional encoding -> feats [65536,40]
    ballq_encode_kernel<<<NCENT / 4, 128, 0, stream>>>(points, cents, feats);
    // 3) layer 1: GEMM + BN + ReLU
    gemm_wmma_kernel<IN_DIM, H1D><<<(NROWS / 16) * (H1D / 16) / 8, 256, 0, stream>>>(feats, W1, b1, y1);
    colsum_partial_kernel<H1D><<<256, 256, 0, stream>>>(y1, psum1, psq1);
    bn_finalize_kernel<H1D><<<1, 128, 0, stream>>>(psum1, psq1, mean1, rstd1);
    bn_relu_kernel<H1D><<<(unsigned)((size_t)NROWS * H1D / 256), 256, 0, stream>>>(y1, g1, be1, mean1, rstd1);
    // 4) layer 2: GEMM + BN + ReLU
    gemm_wmma_kernel<H1D, H2D><<<(NROWS / 16) * (H2D / 16) / 8, 256, 0, stream>>>(y1, W2, b2, y2);
    colsum_partial_kernel<H2D><<<256, 256, 0, stream>>>(y2, psum2, psq2);
    bn_finalize_kernel<H2D><<<1, 128, 0, stream>>>(psum2, psq2, mean2, rstd2);
    bn_relu_kernel<H2D><<<(unsigned)((size_t)NROWS * H2D / 256), 256, 0, stream>>>(y2, g2, be2, mean2, rstd2);
    // 5) layer 3 GEMM fused with maxpool over K=32 -> out [2048,128]
    gemm3_maxpool_kernel<<<NCENT * (OUTD / 16) / 8, 256, 0, stream>>>(y2, W3, b3, out);
}